// RpnTrainingModel_80934363725966
// MI455X (gfx1250) — compile-verified
//
#include <hip/hip_runtime.h>

// ---------------------------------------------------------------------------
// RPN training loss for MI455X (gfx1250, wave32).
// Pipeline:
//   k_count : per-block count of positive anchors (argmax IoU >= 1)
//   k_scan  : exclusive prefix of block counts + num_pos/cur_pos/num_neg
//   k_slots : deterministic first-128 positives / first-256 negatives
//   k_loss  : 256-sample CE + smooth-L1, reduced with V_WMMA_F32_16X16X4_F32
// gt boxes (1.6KB) staged into LDS with the Tensor Data Mover.
// ---------------------------------------------------------------------------

#define GT_MAX        512
#define MAX_POS_K     128
#define TOTAL_SAMPLES 256
#define BLK           256

typedef float        v2f __attribute__((ext_vector_type(2)));
typedef float        v8f __attribute__((ext_vector_type(8)));
typedef unsigned int v4u __attribute__((ext_vector_type(4)));
typedef int          v4i __attribute__((ext_vector_type(4)));
typedef int          v8i __attribute__((ext_vector_type(8)));

#if __has_builtin(__builtin_amdgcn_tensor_load_to_lds) && __has_builtin(__builtin_amdgcn_s_wait_tensorcnt)
#define USE_TDM 1
#else
#define USE_TDM 0
#endif

#if __has_builtin(__builtin_amdgcn_wmma_f32_16x16x4_f32)
#define USE_WMMA 1
#else
#define USE_WMMA 0
#endif

// Stage gt boxes (4*M f32) into LDS via TDM, then compute per-gt areas.
__device__ __forceinline__ void load_gt_to_lds(const float* __restrict__ gtb, int M,
                                               float* shGt, float* shArea) {
  const int n = 4 * M;
#if USE_TDM
  if (threadIdx.x < 32) {  // wave 0 issues the DMA (EXEC ignored by TDM)
    unsigned long long ga = (unsigned long long)gtb;
    // Low 32 bits of a generic LDS pointer are the workgroup-relative LDS byte
    // offset (ISA 10.2).
    unsigned ldsoff = (unsigned)(unsigned long long)(void*)shGt;
    unsigned dim0 = (unsigned)n;  // 1-D tensor of 4*M f32 elements
    // ---- D# group 0: count=1 | lds_addr | global_addr | type=2 ----
    v4u g0;
    g0[0] = 1u;
    g0[1] = ldsoff;
    g0[2] = (unsigned)(ga & 0xFFFFFFFFull);
    g0[3] = (unsigned)((ga >> 32) & 0x01FFFFFFull) | 0x80000000u;
    // ---- D# group 1 ----
    v8i g1;
    g1[0] = (int)(2u << 16);                      // data_size = 4B, no mask/pad
    g1[1] = (int)(dim0 << 16);                    // tensor_dim0[15:0] @bit48
    g1[2] = (int)((dim0 >> 16) | (1u << 16));     // tensor_dim0 hi | tensor_dim1=1
    g1[3] = (int)(dim0 << 16);                    // tensor_dim1 hi=0 | tile_dim0
    g1[4] = 1;                                    // tile_dim1=1, tile_dim2=0
    g1[5] = (int)dim0;                            // tensor_dim0_stride lo32
    g1[6] = (int)((dim0 & 0xFFFFu) << 16);        // stride0 hi=0 | stride1 lo16
    g1[7] = 0;                                    // stride1 hi
    // ---- D# group 2/3: higher dims = 1, strides = dim0, unused tiles = 0 ----
    v4i g2; g2[0] = 1;        g2[1] = 1;         g2[2] = (int)dim0; g2[3] = 0;
    v4i g3; g3[0] = (int)dim0; g3[1] = (1 << 16); g3[2] = 0;         g3[3] = 0;
#if __clang_major__ >= 23
    v8i g4 = {0, 0, 0, 0, 0, 0, 0, 0};
    __builtin_amdgcn_tensor_load_to_lds(g0, g1, g2, g3, g4, 0);
#else
    __builtin_amdgcn_tensor_load_to_lds(g0, g1, g2, g3, 0);
#endif
    __builtin_amdgcn_s_wait_tensorcnt(0);
  }
  __syncthreads();
#else
  for (int t = threadIdx.x; t < n; t += blockDim.x) shGt[t] = gtb[t];
  __syncthreads();
#endif
  for (int j = threadIdx.x; j < M; j += blockDim.x)
    shArea[j] = (shGt[4 * j + 2] - shGt[4 * j + 0]) * (shGt[4 * j + 3] - shGt[4 * j + 1]);
  __syncthreads();
}

// argmax_j IoU(anchor, gt_j); first max wins (matches jnp.argmax tie-break).
__device__ __forceinline__ int argmax_iou(float ax1, float ay1, float ax2, float ay2,
                                          const float* shGt, const float* shArea, int M) {
  float areaA = (ax2 - ax1) * (ay2 - ay1);
  float best = -1.0f;
  int bi = 0;
  for (int j = 0; j < M; ++j) {
    float bx1 = shGt[4 * j + 0], by1 = shGt[4 * j + 1];
    float bx2 = shGt[4 * j + 2], by2 = shGt[4 * j + 3];
    float lx = fmaxf(ax1, bx1), ly = fmaxf(ay1, by1);
    float rx = fminf(ax2, bx2), ry = fminf(ay2, by2);
    float w = fmaxf(rx - lx, 0.0f), h = fmaxf(ry - ly, 0.0f);
    float inter = w * h;
    float iou = inter / (areaA + shArea[j] - inter);
    if (iou > best) { best = iou; bi = j; }
  }
  return bi;
}

__device__ __forceinline__ float smooth_l1(float d) {
  float ad = fabsf(d);
  return (ad < 1.0f) ? 0.5f * d * d : ad - 0.5f;
}

// Exact f32 sum of sh[0..255] using chained V_WMMA_F32_16X16X4_F32 with an
// all-ones B.  Must be called by all 32 lanes of wave 0 (EXEC all-ones).
// A 16x4 f32 layout: lanes 0-15 hold {K0,K1}, lanes 16-31 hold {K2,K3}, row=M.
// D col 0 lives in lane 0 (M=0..7, 8 VGPRs) and lane 16 (M=8..15).
__device__ __forceinline__ float wave0_sum256(const float* sh) {
#if USE_WMMA
  int lane = (int)threadIdx.x;       // 0..31
  int row = lane & 15;
  int kof = (lane >> 4) << 1;        // 0 or 2
  v8f d = {0.f, 0.f, 0.f, 0.f, 0.f, 0.f, 0.f, 0.f};
  v2f ones; ones[0] = 1.0f; ones[1] = 1.0f;
#pragma unroll
  for (int c = 0; c < 4; ++c) {      // 4 chunks of 64 values
    const float* b = sh + (c << 6) + (row << 2) + kof;
    v2f a; a[0] = b[0]; a[1] = b[1];
    d = __builtin_amdgcn_wmma_f32_16x16x4_f32(false, a, false, ones,
                                              (short)0, d, false, false);
  }
  float s = d[0] + d[1] + d[2] + d[3] + d[4] + d[5] + d[6] + d[7];
  return __shfl(s, 0, 32) + __shfl(s, 16, 32);
#else
  float s = 0.0f;
  if (threadIdx.x == 0)
    for (int t = 0; t < TOTAL_SAMPLES; ++t) s += sh[t];
  return s;
#endif
}

// ---------------------------------------------------------------------------

__global__ void k_count(const float* __restrict__ anchors, const float* __restrict__ gtb,
                        int N, int M, int* __restrict__ blockPos) {
  __shared__ float shGt[4 * GT_MAX];
  __shared__ float shArea[GT_MAX];
  __shared__ int wt[BLK / 32];
  load_gt_to_lds(gtb, M, shGt, shArea);
  int i = blockIdx.x * BLK + threadIdx.x;
  int isPos = 0;
  if (i < N) {
    float4 av = reinterpret_cast<const float4*>(anchors)[i];
    isPos = (argmax_iou(av.x, av.y, av.z, av.w, shGt, shArea, M) >= 1) ? 1 : 0;
  }
  unsigned long long m = __ballot(isPos);
  int lane = threadIdx.x & 31, wid = threadIdx.x >> 5;
  if (lane == 0) wt[wid] = __popcll(m);
  __syncthreads();
  if (threadIdx.x == 0) {
    int s = 0;
    for (int w = 0; w < BLK / 32; ++w) s += wt[w];
    blockPos[blockIdx.x] = s;
  }
}

__global__ void k_scan(int* __restrict__ blockPos, int nblocks, int N,
                       int* __restrict__ scalars) {
  if (blockIdx.x == 0 && threadIdx.x == 0) {
    int run = 0;
    for (int b = 0; b < nblocks; ++b) {
      int t = blockPos[b];
      blockPos[b] = run;   // exclusive prefix, in place
      run += t;
    }
    scalars[0] = run;                                    // num_pos
    scalars[1] = (run < MAX_POS_K) ? run : MAX_POS_K;    // cur_pos
    scalars[2] = N - run;                                // num_neg
  }
}

__global__ void k_slots(const float* __restrict__ anchors, const float* __restrict__ gtb,
                        int N, int M, const int* __restrict__ blockPosPrefix,
                        const int* __restrict__ scalars,
                        int* __restrict__ posSlots, int* __restrict__ negSlots) {
  __shared__ float shGt[4 * GT_MAX];
  __shared__ float shArea[GT_MAX];
  __shared__ int wt[BLK / 32];
  load_gt_to_lds(gtb, M, shGt, shArea);
  int i = blockIdx.x * BLK + threadIdx.x;
  int isPos = 0;
  if (i < N) {
    float4 av = reinterpret_cast<const float4*>(anchors)[i];
    isPos = (argmax_iou(av.x, av.y, av.z, av.w, shGt, shArea, M) >= 1) ? 1 : 0;
  }
  unsigned long long m = __ballot(isPos);
  int lane = threadIdx.x & 31, wid = threadIdx.x >> 5;
  int lp = __popcll(m & ((1ull << lane) - 1ull));
  if (lane == 0) wt[wid] = __popcll(m);
  __syncthreads();
  int base = 0;
  for (int w = 0; w < wid; ++w) base += wt[w];
  if (i < N) {
    int posBefore = blockPosPrefix[blockIdx.x] + base + lp;  // #pos with idx < i
    if (isPos) {
      if (posBefore < MAX_POS_K) posSlots[posBefore] = i;
      int comb = scalars[2] + posBefore;      // neg_order tail = positives
      if (comb < TOTAL_SAMPLES) negSlots[comb] = i;
    } else {
      int negBefore = i - posBefore;          // all anchors are labeled 0/1
      if (negBefore < TOTAL_SAMPLES) negSlots[negBefore] = i;
    }
  }
}

__global__ void k_loss(const float* __restrict__ score, const float* __restrict__ pred,
                       const float* __restrict__ anchors, const float* __restrict__ gtb,
                       int M, const int* __restrict__ scalars,
                       const int* __restrict__ posSlots, const int* __restrict__ negSlots,
                       float* __restrict__ out) {
  __shared__ float shGt[4 * GT_MAX];
  __shared__ float shArea[GT_MAX];
  __shared__ float red[2 * TOTAL_SAMPLES];
  load_gt_to_lds(gtb, M, shGt, shArea);
  int i = (int)threadIdx.x;  // 0..255
  int cur_pos = scalars[1];
  int sel = (i < cur_pos) ? posSlots[i] : negSlots[i - cur_pos];
  float4 av = reinterpret_cast<const float4*>(anchors)[sel];
  int g = argmax_iou(av.x, av.y, av.z, av.w, shGt, shArea, M);
  int label = (g >= 1) ? 1 : 0;
  // classification: -log_softmax(score[sel])[label]
  float2 sc = reinterpret_cast<const float2*>(score)[sel];
  float mx = fmaxf(sc.x, sc.y);
  float lse = mx + logf(expf(sc.x - mx) + expf(sc.y - mx));
  float cls = lse - (label ? sc.y : sc.x);
  // regression: REG_W * sum smooth_l1(label * (pred - encode(anchor, gt)))
  float reg = 0.0f;
  if (label) {
    float aw = av.z - av.x, ah = av.w - av.y;
    float acx = av.x + 0.5f * aw, acy = av.y + 0.5f * ah;
    float gx1 = shGt[4 * g + 0], gy1 = shGt[4 * g + 1];
    float gx2 = shGt[4 * g + 2], gy2 = shGt[4 * g + 3];
    float gw = gx2 - gx1, gh = gy2 - gy1;
    float gcx = gx1 + 0.5f * gw, gcy = gy1 + 0.5f * gh;
    float t0 = (gcx - acx) / aw, t1 = (gcy - acy) / ah;
    float t2 = logf(gw / aw), t3 = logf(gh / ah);
    float4 p = reinterpret_cast<const float4*>(pred)[sel];
    reg = smooth_l1(p.x - t0) + smooth_l1(p.y - t1) +
          smooth_l1(p.z - t2) + smooth_l1(p.w - t3);
    reg *= 2.0f;  // REG_W
  }
  red[i] = cls;
  red[TOTAL_SAMPLES + i] = reg;
  __syncthreads();
  if (threadIdx.x < 32) {  // wave 0 fully active -> EXEC all ones for WMMA
    float cs = wave0_sum256(red);
    float rs = wave0_sum256(red + TOTAL_SAMPLES);
    if (threadIdx.x == 0) {
      out[0] = cs * (1.0f / TOTAL_SAMPLES);  // CLS_W = 1
      out[1] = rs * (1.0f / TOTAL_SAMPLES);
    }
  }
}

// ---------------------------------------------------------------------------

extern "C" void kernel_launch(void* const* d_in, const int* in_sizes, int n_in,
                              void* d_out, int out_size, void* d_ws, size_t ws_size,
                              hipStream_t stream) {
  const float* score   = (const float*)d_in[0];  // [N,2]
  const float* pred    = (const float*)d_in[1];  // [N,4]
  const float* anchors = (const float*)d_in[2];  // [N,4]
  const float* gtb     = (const float*)d_in[3];  // [M,4]
  int N = in_sizes[2] / 4;
  int M = in_sizes[3] / 4;
  if (M > GT_MAX) M = GT_MAX;
  int nblocks = (N + BLK - 1) / BLK;

  int* blockPos = (int*)d_ws;            // [nblocks] counts -> prefixes
  int* scalars  = blockPos + nblocks;    // [4]: num_pos, cur_pos, num_neg
  int* posSlots = scalars + 4;           // [128]
  int* negSlots = posSlots + MAX_POS_K;  // [256]

  k_count<<<nblocks, BLK, 0, stream>>>(anchors, gtb, N, M, blockPos);
  k_scan<<<1, 1, 0, stream>>>(blockPos, nblocks, N, scalars);
  k_slots<<<nblocks, BLK, 0, stream>>>(anchors, gtb, N, M, blockPos, scalars,
                                       posSlots, negSlots);
  k_loss<<<1, TOTAL_SAMPLES, 0, stream>>>(score, pred, anchors, gtb, M, scalars,
                                          posSlots, negSlots, (float*)d_out);
}